// KANLayer_8504035246521
// MI455X (gfx1250) — compile-verified
//
#include <hip/hip_runtime.h>
#include <stdint.h>

// ---------------- problem constants ----------------
#define BATCH   64
#define D_IN    512
#define D_OUT   512
#define GRIDN   100
#define KDEG    4          // DEG+1
#define LN_EPS   1e-5f
#define GRID_EPS 1e-6f

#define QSPLIT  4          // i-parallel split within a block
#define STEPS   (D_IN / QSPLIT)   // 128 gather steps per thread
#define DEPTH   8          // async LDS ring depth (issue-ahead = DEPTH-1)

typedef __attribute__((ext_vector_type(2))) float v2f;
typedef __attribute__((ext_vector_type(8))) float v8f;

// ---------------- CDNA5 feature probes (host pass falls back cleanly) ------
#if __has_builtin(__builtin_amdgcn_wmma_f32_16x16x4_f32)
#define HAVE_WMMA 1
#else
#define HAVE_WMMA 0
#endif

#if __has_builtin(__builtin_amdgcn_global_load_async_to_lds_b128) && \
    __has_builtin(__builtin_amdgcn_s_wait_asynccnt)
#define USE_ASYNC 1
#else
#define USE_ASYNC 0
#endif

#if USE_ASYNC
// Exact parameter types per hipcc diagnostic: 'int __vector(4)' pointers,
// global side AS1, LDS side AS3.
typedef int vint4 __attribute__((vector_size(16)));
typedef __attribute__((address_space(1))) vint4* gas_b128_t;
typedef __attribute__((address_space(3))) vint4* las_b128_t;

__device__ __forceinline__ void async_ld_b128(const void* g, void* l) {
  // Per-lane gathered global addr -> per-lane LDS slot, tracked by ASYNCcnt.
  __builtin_amdgcn_global_load_async_to_lds_b128(
      (gas_b128_t)(g), (las_b128_t)(l),
      0 /*ioffset*/, 0 /*cpol: RT so L2 keeps cross-batch cacheline reuse*/);
}
#endif

// =====================================================================
// Fused KAN layer kernel.
//   Phase 1 (per block, redundant across the 8 o-tiles of a batch row):
//     LayerNorm -> grid index + frac t; Bernstein basis via
//     V_WMMA_F32_16X16X4_F32 (power_bases Nx4 @ basis_matrix 4x4), beta -> LDS.
//   Phase 2: out[b,o] = sum_i dot4(beta[i,:], P[i,o,idx[i],:])
//     256 threads = 64 o x 4-way i-split; each thread's 16B gathers are
//     pipelined through an 8-deep LDS ring with global_load_async_to_lds_b128
//     (ASYNCcnt, per-wave, no barriers in the steady state).
// Roofline: ~419 MB of P traffic @ 23.3 TB/s ~= 18 us; everything else is noise.
// =====================================================================
__global__ __launch_bounds__(256) void kan_fused(
    const float* __restrict__ x, const float* __restrict__ lnw,
    const float* __restrict__ lnb, const float* __restrict__ Bm,
    const float* __restrict__ P, float* __restrict__ out)
{
  const int b   = blockIdx.y;          // batch row
  const int ot  = blockIdx.x;          // o-tile 0..7
  const int tid = threadIdx.x;

  __shared__ float red[256];
  __shared__ float t_sh[D_IN];
  __shared__ int   gi_sh[D_IN];
  __shared__ float beta_sh[D_IN * KDEG];   // 8 KB
#if USE_ASYNC
  __shared__ float pbuf[DEPTH][256 * 4];   // 32 KB async gather ring
#endif

  // ---------------- Phase 1a: LayerNorm statistics ----------------
  const float x0 = x[b * D_IN + tid];
  const float x1 = x[b * D_IN + 256 + tid];

  red[tid] = x0 + x1;
  __syncthreads();
  for (int off = 128; off > 0; off >>= 1) {
    if (tid < off) red[tid] += red[tid + off];
    __syncthreads();
  }
  const float mean = red[0] * (1.0f / (float)D_IN);
  __syncthreads();

  const float d0 = x0 - mean, d1 = x1 - mean;
  red[tid] = d0 * d0 + d1 * d1;
  __syncthreads();
  for (int off = 128; off > 0; off >>= 1) {
    if (tid < off) red[tid] += red[tid + off];
    __syncthreads();
  }
  const float rstd = rsqrtf(red[0] * (1.0f / (float)D_IN) + LN_EPS);
  __syncthreads();

  // ---------------- Phase 1b: normalize, clip, (grid index, frac) ----------
  #pragma unroll
  for (int rep = 0; rep < 2; ++rep) {
    const int   e  = tid + rep * 256;
    const float xv = rep ? x1 : x0;
    float xn = (xv - mean) * rstd * lnw[e] + lnb[e];
    xn = fminf(fmaxf(xn, -1.0f + GRID_EPS), 1.0f - GRID_EPS);
    const float scaled = (xn + 1.0f) * 0.5f * (float)GRIDN;
    int gi = (int)floorf(scaled);
    gi = gi < 0 ? 0 : (gi > GRIDN - 1 ? GRIDN - 1 : gi);
    t_sh[e]  = scaled - (float)gi;
    gi_sh[e] = gi;
  }
  __syncthreads();

  // ---------------- Phase 1c: Bernstein basis (WMMA) -> beta_sh ------------
#if HAVE_WMMA
  {
    const int  wv   = tid >> 5;        // wave id (8 waves)
    const int  lane = tid & 31;
    const int  n    = lane & 15;
    const bool hi   = lane >= 16;

    // B fragment: 4x16 f32, rows K0/K1 in vgpr0 (lo/hi lanes), K2/K3 in vgpr1.
    v2f bfrag;
    bfrag.x = (n < KDEG) ? Bm[(hi ? 1 : 0) * KDEG + n] : 0.0f;
    bfrag.y = (n < KDEG) ? Bm[(hi ? 3 : 2) * KDEG + n] : 0.0f;

    // Each wave handles 64 elements: 4 WMMAs of 16 rows each.
    #pragma unroll
    for (int j = 0; j < 4; ++j) {
      const int   e0 = wv * 64 + j * 16;
      const float t  = t_sh[e0 + n];
      const float t2 = t * t;
      // A fragment 16x4 f32: vgpr0 = {K0 | K2}, vgpr1 = {K1 | K3}
      v2f afrag;
      afrag.x = hi ? t2 : 1.0f;
      afrag.y = hi ? t2 * t : t;
      v8f c = {0.f, 0.f, 0.f, 0.f, 0.f, 0.f, 0.f, 0.f};
      v8f d = __builtin_amdgcn_wmma_f32_16x16x4_f32(
          false, afrag, false, bfrag, (short)0, c, false, false);
      if (n < KDEG) {
        #pragma unroll
        for (int r = 0; r < 8; ++r) {
          const int m = r + (hi ? 8 : 0);   // D: vgpr r holds M=r (lo) / r+8 (hi)
          beta_sh[(e0 + m) * KDEG + n] = d[r];
        }
      }
    }
  }
#else
  #pragma unroll
  for (int rep = 0; rep < 2; ++rep) {
    const int   e = tid + rep * 256;
    const float t = t_sh[e];
    const float p0 = 1.0f, p1 = t, p2 = t * t, p3 = t * t * t;
    #pragma unroll
    for (int n = 0; n < KDEG; ++n)
      beta_sh[e * KDEG + n] =
          p0 * Bm[n] + p1 * Bm[4 + n] + p2 * Bm[8 + n] + p3 * Bm[12 + n];
  }
#endif
  __syncthreads();

  // ---------------- Phase 2: pipelined gather-accumulate -------------------
  const int oo = tid & 63;
  const int q  = tid >> 6;             // 0..3 : i-parallel split
  const int o  = ot * 64 + oo;
  const char* gbase = (const char*)P;
  float acc = 0.0f;

#if USE_ASYNC
  // byte offset of P[i][o][gi][0] = ((i*512 + o)*100 + gi)*16 ; max ~419MB < 2^32
  uint32_t bb = (uint32_t)(q * D_OUT + o) * (uint32_t)(GRIDN * 16);
  const uint32_t step = (uint32_t)(QSPLIT * D_OUT) * (uint32_t)(GRIDN * 16);

  #pragma unroll
  for (int s = 0; s < DEPTH - 1; ++s) {          // prologue: issue-ahead 7
    const int i = q + s * QSPLIT;
    async_ld_b128(gbase + (size_t)(bb + (uint32_t)gi_sh[i] * 16u), &pbuf[s][tid * 4]);
    bb += step;
  }

  for (int s = 0; s < STEPS; ++s) {
    const int i = q + s * QSPLIT;
    if (s + DEPTH - 1 < STEPS) {
      // ring slot (s+DEPTH-1)&(DEPTH-1) was last read at iteration s-1:
      // make sure that ds_load retired before the async engine overwrites it.
      asm volatile("s_wait_dscnt 0" ::: "memory");
      const int ipf = i + (DEPTH - 1) * QSPLIT;
      async_ld_b128(gbase + (size_t)(bb + (uint32_t)gi_sh[ipf] * 16u),
                    &pbuf[(s + DEPTH - 1) & (DEPTH - 1)][tid * 4]);
      bb += step;
      __builtin_amdgcn_s_wait_asynccnt(DEPTH - 1);   // slot s's gather done
    } else {
      __builtin_amdgcn_s_wait_asynccnt(0);           // drain tail
    }
    asm volatile("" ::: "memory");                   // keep LDS reads below wait

    const float4 p  = *(const float4*)&pbuf[s & (DEPTH - 1)][tid * 4];
    const float4 be = *(const float4*)&beta_sh[i * 4];   // wave-uniform broadcast
    acc = fmaf(p.x, be.x, fmaf(p.y, be.y, fmaf(p.z, be.z, fmaf(p.w, be.w, acc))));
  }
#else
  for (int s = 0; s < STEPS; ++s) {
    const int i = q + s * QSPLIT;
    const size_t off = ((size_t)(i * D_OUT + o) * GRIDN + (size_t)gi_sh[i]) * 16u;
    const float4 p  = *(const float4*)(gbase + off);
    const float4 be = *(const float4*)&beta_sh[i * 4];
    acc = fmaf(p.x, be.x, fmaf(p.y, be.y, fmaf(p.z, be.z, fmaf(p.w, be.w, acc))));
  }
#endif

  // ---------------- Phase 3: 4-way i-split reduction, store ----------------
  red[tid] = acc;
  __syncthreads();
  if (tid < 64) {
    out[b * D_OUT + ot * 64 + tid] =
        red[tid] + red[tid + 64] + red[tid + 128] + red[tid + 192];
  }
}

// =====================================================================
extern "C" void kernel_launch(void* const* d_in, const int* in_sizes, int n_in,
                              void* d_out, int out_size, void* d_ws, size_t ws_size,
                              hipStream_t stream) {
  (void)in_sizes; (void)n_in; (void)out_size; (void)d_ws; (void)ws_size;
  const float* x   = (const float*)d_in[0];
  const float* P   = (const float*)d_in[1];
  const float* lnw = (const float*)d_in[2];
  const float* lnb = (const float*)d_in[3];
  const float* Bm  = (const float*)d_in[4];
  float* out = (float*)d_out;

  kan_fused<<<dim3(8, BATCH), 256, 0, stream>>>(x, lnw, lnb, Bm, P, out);
}